// MAF_Extractor_15564961481181
// MI455X (gfx1250) — compile-verified
//
#include <hip/hip_runtime.h>

typedef float v2f __attribute__((ext_vector_type(2)));
typedef float v8f __attribute__((ext_vector_type(8)));

#define LEAK 0.01f

// Fused: projection + bilinear grid-sample (to LDS) + 3-layer MLP via
// V_WMMA_F32_16X16X4_F32 (f32 A/B, f32 acc — matches reference precision).
// One workgroup = one batch x 16-point tile. 256 threads = 8 wave32.
__launch_bounds__(256, 2)
__global__ void maf_fused_kernel(
    const float* __restrict__ P,     // (B,N,3)
    const float* __restrict__ CAM,   // (B,3): s, tx, ty
    const float* __restrict__ SF,    // (B,256,56,56)
    const float* __restrict__ W0, const float* __restrict__ B0,   // (128,256)
    const float* __restrict__ W1, const float* __restrict__ B1,   // (64,384)
    const float* __restrict__ W2, const float* __restrict__ B2,   // (5,320)
    float* __restrict__ OUT,         // (B, 5*N)
    int N)
{
    constexpr int C  = 256;
    constexpr int H  = 56;
    constexpr int Wd = 56;
    constexpr int D0 = 128;
    constexpr int D1 = 64;
    constexpr int D2 = 5;

    // K-paired B-matrix layouts: elem (k, n) at [(k>>1)*32 + n*2 + (k&1)]
    // -> per-lane WMMA B fragment (K=(lane>>4)*2+{0,1}, N=lane&15) is one
    //    8B-aligned ds_load_b64, conflict-free across the wave.
    __shared__ float featB[C  * 16];   // 16 KB
    __shared__ float y0B [D0 * 16];    //  8 KB
    __shared__ float y1B [D1 * 16];    //  4 KB
    __shared__ int   pX[16][2];
    __shared__ int   pY[16][2];
    __shared__ float pWt[16][4];       // w00, w10, w01, w11 (validity folded)

    const int tid  = threadIdx.x;
    const int b    = blockIdx.y;
    const int tile = blockIdx.x;

    // ---------- stage 1: projection + bilinear setup (16 threads) ----------
    if (tid < 16) {
        const int n    = tile * 16 + tid;
        const bool live = (n < N);
        const int nn   = live ? n : 0;
        const float s  = CAM[b * 3 + 0];
        const float tx = CAM[b * 3 + 1];
        const float ty = CAM[b * 3 + 2];
        const float px = s * (P[((size_t)b * N + nn) * 3 + 0] + tx);
        const float py = s * (P[((size_t)b * N + nn) * 3 + 1] + ty);
        const float x  = (px + 1.0f) * 0.5f * (float)(Wd - 1);
        const float y  = (py + 1.0f) * 0.5f * (float)(H  - 1);
        const float xf = floorf(x), yf = floorf(y);
        const float wx1 = x - xf, wx0 = 1.0f - wx1;
        const float wy1 = y - yf, wy0 = 1.0f - wy1;
        const int x0i = (int)xf, y0i = (int)yf;
        const int x1i = x0i + 1, y1i = y0i + 1;
        const float vx0 = (x0i >= 0 && x0i < Wd) ? 1.0f : 0.0f;
        const float vx1 = (x1i >= 0 && x1i < Wd) ? 1.0f : 0.0f;
        const float vy0 = (y0i >= 0 && y0i < H ) ? 1.0f : 0.0f;
        const float vy1 = (y1i >= 0 && y1i < H ) ? 1.0f : 0.0f;
        const float liv = live ? 1.0f : 0.0f;
        pWt[tid][0] = wx0 * wy0 * vx0 * vy0 * liv;
        pWt[tid][1] = wx1 * wy0 * vx1 * vy0 * liv;
        pWt[tid][2] = wx0 * wy1 * vx0 * vy1 * liv;
        pWt[tid][3] = wx1 * wy1 * vx1 * vy1 * liv;
        pX[tid][0] = min(max(x0i, 0), Wd - 1);
        pX[tid][1] = min(max(x1i, 0), Wd - 1);
        pY[tid][0] = min(max(y0i, 0), H  - 1);
        pY[tid][1] = min(max(y1i, 0), H  - 1);
    }
    __syncthreads();

    // ---------- stage 2: gather 256ch x 16pts into LDS (all 256 thr) ------
    const float* fplane = SF + (size_t)b * C * H * Wd;
    for (int idx = tid; idx < C * 16; idx += 256) {
        const int c  = idx >> 4;
        const int pt = idx & 15;
        const float* cp = fplane + (size_t)c * (H * Wd);
        const int r0 = pY[pt][0] * Wd, r1 = pY[pt][1] * Wd;
        const int x0 = pX[pt][0],      x1 = pX[pt][1];
        const float v = pWt[pt][0] * cp[r0 + x0]
                      + pWt[pt][1] * cp[r0 + x1]
                      + pWt[pt][2] * cp[r1 + x0]
                      + pWt[pt][3] * cp[r1 + x1];
        featB[(c >> 1) * 32 + pt * 2 + (c & 1)] = v;
    }
    __syncthreads();

    const int wave  = tid >> 5;       // wave32
    const int lane  = tid & 31;
    const int col   = lane & 15;      // A row-in-tile / B-N col / D col
    const int khalf = lane >> 4;      // K-half select for A/B fragments

    // ---------- layer 0: y0 = lrelu(W0 @ feat + b0), 8 waves x 16 rows ----
    {
        const int rm   = wave * 16;
        const int rowA = rm + col;
        v8f acc = {0.f, 0.f, 0.f, 0.f, 0.f, 0.f, 0.f, 0.f};
        for (int k = 0; k < C; k += 4) {
            const int ka = k + khalf * 2;
            v2f a  = *(const v2f*)(W0 + (size_t)rowA * C + ka);
            v2f bb = *(const v2f*)(featB + (ka >> 1) * 32 + col * 2);
            acc = __builtin_amdgcn_wmma_f32_16x16x4_f32(
                false, a, false, bb, (short)0, acc, false, false);
        }
        for (int v = 0; v < 8; ++v) {
            const int r = rm + v + khalf * 8;
            float val = acc[v] + B0[r];
            val = (val > 0.0f) ? val : LEAK * val;
            y0B[(r >> 1) * 32 + col * 2 + (r & 1)] = val;
        }
    }
    __syncthreads();

    // ---------- layer 1: y1 = lrelu(W1 @ [y0;feat] + b1), waves 0..3 ------
    if (wave < 4) {
        const int rm   = wave * 16;
        const int rowA = rm + col;
        v8f acc = {0.f, 0.f, 0.f, 0.f, 0.f, 0.f, 0.f, 0.f};
        for (int k = 0; k < D0; k += 4) {               // K-slab 1: y0
            const int ka = k + khalf * 2;
            v2f a  = *(const v2f*)(W1 + (size_t)rowA * (D0 + C) + ka);
            v2f bb = *(const v2f*)(y0B + (ka >> 1) * 32 + col * 2);
            acc = __builtin_amdgcn_wmma_f32_16x16x4_f32(
                false, a, false, bb, (short)0, acc, false, false);
        }
        for (int k = 0; k < C; k += 4) {                // K-slab 2: feat
            const int ka = k + khalf * 2;
            v2f a  = *(const v2f*)(W1 + (size_t)rowA * (D0 + C) + D0 + ka);
            v2f bb = *(const v2f*)(featB + (ka >> 1) * 32 + col * 2);
            acc = __builtin_amdgcn_wmma_f32_16x16x4_f32(
                false, a, false, bb, (short)0, acc, false, false);
        }
        for (int v = 0; v < 8; ++v) {
            const int r = rm + v + khalf * 8;
            float val = acc[v] + B1[r];
            val = (val > 0.0f) ? val : LEAK * val;
            y1B[(r >> 1) * 32 + col * 2 + (r & 1)] = val;
        }
    }
    __syncthreads();

    // ---------- layer 2: out = relu(W2 @ [y1;feat] + b2), wave 0 ----------
    if (wave == 0) {
        v8f acc = {0.f, 0.f, 0.f, 0.f, 0.f, 0.f, 0.f, 0.f};
        for (int k = 0; k < D1; k += 4) {               // K-slab 1: y1
            const int ka = k + khalf * 2;
            v2f a = {0.f, 0.f};
            if (col < D2) a = *(const v2f*)(W2 + (size_t)col * (D1 + C) + ka);
            v2f bb = *(const v2f*)(y1B + (ka >> 1) * 32 + col * 2);
            acc = __builtin_amdgcn_wmma_f32_16x16x4_f32(
                false, a, false, bb, (short)0, acc, false, false);
        }
        for (int k = 0; k < C; k += 4) {                // K-slab 2: feat
            const int ka = k + khalf * 2;
            v2f a = {0.f, 0.f};
            if (col < D2) a = *(const v2f*)(W2 + (size_t)col * (D1 + C) + D1 + ka);
            v2f bb = *(const v2f*)(featB + (ka >> 1) * 32 + col * 2);
            acc = __builtin_amdgcn_wmma_f32_16x16x4_f32(
                false, a, false, bb, (short)0, acc, false, false);
        }
        const int n = tile * 16 + col;
        if (n < N) {
            for (int v = 0; v < 8; ++v) {
                const int r = v + khalf * 8;            // output channel
                if (r < D2) {
                    float val = acc[v] + B2[r];
                    val = (val > 0.0f) ? val : 0.0f;
                    OUT[(size_t)b * (D2 * N) + (size_t)r * N + n] = val;
                }
            }
        }
    }
}

extern "C" void kernel_launch(void* const* d_in, const int* in_sizes, int n_in,
                              void* d_out, int out_size, void* d_ws, size_t ws_size,
                              hipStream_t stream) {
    (void)n_in; (void)out_size; (void)d_ws; (void)ws_size;
    const float* P   = (const float*)d_in[0];
    const float* CAM = (const float*)d_in[1];
    const float* SF  = (const float*)d_in[2];
    const float* W0  = (const float*)d_in[3];
    const float* B0  = (const float*)d_in[4];
    const float* W1  = (const float*)d_in[5];
    const float* B1  = (const float*)d_in[6];
    const float* W2  = (const float*)d_in[7];
    const float* B2  = (const float*)d_in[8];
    float* OUT = (float*)d_out;

    const int B = in_sizes[1] / 3;            // cam is (B,3)
    const int N = in_sizes[0] / (3 * B);      // p is (B,N,3)

    dim3 grid((N + 15) / 16, B);
    maf_fused_kernel<<<grid, 256, 0, stream>>>(P, CAM, SF, W0, B0, W1, B1,
                                               W2, B2, OUT, N);
}